// LightGCN_32658931319096
// MI455X (gfx1250) — compile-verified
//
#include <hip/hip_runtime.h>
#include <hip/hip_bf16.h>

#define N_USERS 200000
#define N_ITEMS 100000
#define NNODES  (N_USERS + N_ITEMS)   // 300000, divisible by 16
#define EDIM    64
#define NEDGES  1200000
#define SCAN_B  256
#define NBLK_SCAN ((NNODES + SCAN_B - 1) / SCAN_B)   // 1172

typedef __attribute__((ext_vector_type(2))) float v2f;
typedef __attribute__((ext_vector_type(8))) float v8f;

// ---------------- degree histogram (int) ----------------
__global__ void lg_deg_kernel(const int* __restrict__ row, int* __restrict__ cnt) {
    int e = blockIdx.x * blockDim.x + threadIdx.x;
    if (e < NEDGES) atomicAdd(&cnt[row[e]], 1);
}

// ---------------- dis[i] = deg>0 ? rsqrt(deg) : 0 ----------------
__global__ void lg_dis_kernel(const int* __restrict__ cnt, float* __restrict__ dis) {
    int i = blockIdx.x * blockDim.x + threadIdx.x;
    if (i < NNODES) {
        int d = cnt[i];
        dis[i] = (d > 0) ? rsqrtf((float)d) : 0.0f;
    }
}

// ---------------- exclusive scan, stage 1: per-block scan in LDS ----------------
__global__ void lg_scan_block(const int* __restrict__ cnt, int* __restrict__ rptr,
                              int* __restrict__ blksum) {
    __shared__ int sm[SCAN_B];
    int i = blockIdx.x * SCAN_B + threadIdx.x;
    int v = (i < NNODES) ? cnt[i] : 0;
    sm[threadIdx.x] = v;
    __syncthreads();
    for (int off = 1; off < SCAN_B; off <<= 1) {        // Hillis-Steele inclusive scan
        int t = (threadIdx.x >= off) ? sm[threadIdx.x - off] : 0;
        __syncthreads();
        sm[threadIdx.x] += t;
        __syncthreads();
    }
    if (i < NNODES) rptr[i] = sm[threadIdx.x] - v;      // exclusive
    if (threadIdx.x == SCAN_B - 1) blksum[blockIdx.x] = sm[SCAN_B - 1];
}

// ---------------- exclusive scan, stage 2: scan of 1172 block sums ----------------
__global__ void lg_scan_blksums(int* __restrict__ blksum) {
    if (blockIdx.x == 0 && threadIdx.x == 0) {
        int running = 0;
        for (int b = 0; b < NBLK_SCAN; ++b) {
            int t = blksum[b];
            blksum[b] = running;
            running += t;
        }
    }
}

// ---------------- exclusive scan, stage 3: add block offsets ----------------
__global__ void lg_scan_add(int* __restrict__ rptr, const int* __restrict__ blkoff) {
    int i = blockIdx.x * SCAN_B + threadIdx.x;          // same grid shape as stage 1
    if (i < NNODES) rptr[i] += blkoff[blockIdx.x];
    if (i == 0) rptr[NNODES] = NEDGES;
}

// ---------------- CSR fill: slot via per-row atomic counter ----------------
__global__ void lg_fill_kernel(const int* __restrict__ row, const int* __restrict__ col,
                               const float* __restrict__ dis, const int* __restrict__ rptr,
                               int* __restrict__ fill, int* __restrict__ ccol,
                               float* __restrict__ cw) {
    int e = blockIdx.x * blockDim.x + threadIdx.x;
    if (e >= NEDGES) return;
    __builtin_prefetch(row + e + 4096, 0, 1);           // stream-ahead (global_prefetch_b8)
    int r = row[e], c = col[e];
    int pos = rptr[r] + atomicAdd(&fill[r], 1);
    ccol[pos] = c;
    cw[pos]   = dis[c];                                 // dis[row] factored out of the sum
}

// ---------------- h0 = acc = concat(user, item), float4 vectorized ----------------
__global__ void lg_init_kernel(const float4* __restrict__ user, const float4* __restrict__ item,
                               float4* __restrict__ h, float4* __restrict__ acc) {
    int i = blockIdx.x * blockDim.x + threadIdx.x;      // over NNODES*16 float4s
    if (i < NNODES * (EDIM / 4)) {
        float4 v = (i < N_USERS * (EDIM / 4)) ? user[i] : item[i - N_USERS * (EDIM / 4)];
        h[i]   = v;
        acc[i] = v;
    }
}

// ---------------- gather SpMM: 16 threads/node, no atomics; optional acc += h ----------------
template <bool ACC>
__global__ void lg_gather_kernel(const int* __restrict__ rptr, const int* __restrict__ ccol,
                                 const float* __restrict__ cw, const float* __restrict__ dis,
                                 const float4* __restrict__ hin, float4* __restrict__ hout,
                                 float4* __restrict__ acc) {
    int t = blockIdx.x * blockDim.x + threadIdx.x;      // NNODES*16 threads
    int r = t >> 4;
    int c = t & 15;                                     // float4 chunk 0..15
    if (r >= NNODES) return;
    int k0 = rptr[r], k1 = rptr[r + 1];
    float4 s = make_float4(0.f, 0.f, 0.f, 0.f);
    for (int k = k0; k < k1; ++k) {
        float  w = cw[k];                               // broadcast across the 16 chunk threads
        float4 v = hin[(size_t)ccol[k] * (EDIM / 4) + c]; // random 256B row gather (L2-resident)
        s.x += w * v.x; s.y += w * v.y; s.z += w * v.z; s.w += w * v.w;
    }
    float dr = dis[r];                                  // apply row factor once
    s.x *= dr; s.y *= dr; s.z *= dr; s.w *= dr;
    size_t o = (size_t)r * (EDIM / 4) + c;
    hout[o] = s;
    if (ACC) {
        float4 a = acc[o];
        a.x += s.x; a.y += s.y; a.z += s.z; a.w += s.w;
        acc[o] = a;
    }
}

// ---------------- WMMA epilogue: out = 0.25*(acc + h3) on the matrix pipe ----------------
// One wave handles 16 rows x 64 cols. Per 16x16 tile: D = sum_j Sj(16x4) x Bj(4x16) + C with
// S = 0.25*I, chained over the acc tile then the h3 tile -> D = 0.25*(acc+h3), exact in fp32.
__global__ void lg_finalize_wmma(const float* __restrict__ acc, const float* __restrict__ h3,
                                 float* __restrict__ out) {
    int wave = blockIdx.x * (blockDim.x >> 5) + (threadIdx.x >> 5);
    int row0 = wave * 16;
    if (row0 >= NNODES) return;                         // wave-uniform: EXEC stays all-ones
    int lane = threadIdx.x & 31;
    int m    = lane & 15;                               // matrix row (A) / col (B,C,D) index
    int half = lane >> 4;

    for (int col0 = 0; col0 < EDIM; col0 += 16) {
        const float* pa = acc + (size_t)row0 * EDIM + col0 + m;
        const float* ph = h3  + (size_t)row0 * EDIM + col0 + m;
        v8f cacc = {};
#pragma unroll
        for (int j = 0; j < 4; ++j) {                   // 0.25*I x acc_tile
            int r0 = 4 * j + 2 * half;
            v2f a; a.x = (m == r0) ? 0.25f : 0.0f; a.y = (m == r0 + 1) ? 0.25f : 0.0f;
            v2f b; b.x = pa[(size_t)r0 * EDIM]; b.y = pa[(size_t)(r0 + 1) * EDIM];
            cacc = __builtin_amdgcn_wmma_f32_16x16x4_f32(false, a, false, b,
                                                         (short)0, cacc, false, false);
        }
#pragma unroll
        for (int j = 0; j < 4; ++j) {                   // += 0.25*I x h3_tile
            int r0 = 4 * j + 2 * half;
            v2f a; a.x = (m == r0) ? 0.25f : 0.0f; a.y = (m == r0 + 1) ? 0.25f : 0.0f;
            v2f b; b.x = ph[(size_t)r0 * EDIM]; b.y = ph[(size_t)(r0 + 1) * EDIM];
            cacc = __builtin_amdgcn_wmma_f32_16x16x4_f32(false, a, false, b,
                                                         (short)0, cacc, false, false);
        }
#pragma unroll
        for (int v = 0; v < 8; ++v)                     // C/D layout: VGPR v -> row v + 8*half
            out[(size_t)(row0 + v + 8 * half) * EDIM + col0 + m] = cacc[v];
    }
}

static inline size_t align256(size_t x) { return (x + 255) & ~(size_t)255; }

extern "C" void kernel_launch(void* const* d_in, const int* in_sizes, int n_in,
                              void* d_out, int out_size, void* d_ws, size_t ws_size,
                              hipStream_t stream) {
    const float* user = (const float*)d_in[0];
    const float* item = (const float*)d_in[1];
    const int*   row  = (const int*)d_in[2];            // edge_index[0]
    const int*   col  = row + NEDGES;                   // edge_index[1]
    float*       out  = (float*)d_out;                  // [NNODES*EDIM], also the accumulator

    char* ws = (char*)d_ws;
    size_t o_cnt  = 0;                                           // N ints (deg, then fill ctr)
    size_t o_dis  = align256(o_cnt  + (size_t)NNODES * 4);       // N floats
    size_t o_rptr = align256(o_dis  + (size_t)NNODES * 4);       // N+1 ints
    size_t o_blk  = align256(o_rptr + (size_t)(NNODES + 1) * 4); // block sums
    size_t o_ccol = align256(o_blk  + (size_t)NBLK_SCAN * 4);    // E ints
    size_t o_cw   = align256(o_ccol + (size_t)NEDGES * 4);       // E floats
    size_t o_hA   = align256(o_cw   + (size_t)NEDGES * 4);       // N*64 floats
    size_t o_hB   = align256(o_hA   + (size_t)NNODES * EDIM * 4);
    int*   cnt  = (int*)(ws + o_cnt);
    float* dis  = (float*)(ws + o_dis);
    int*   rptr = (int*)(ws + o_rptr);
    int*   blk  = (int*)(ws + o_blk);
    int*   ccol = (int*)(ws + o_ccol);
    float* cw   = (float*)(ws + o_cw);
    float* hA   = (float*)(ws + o_hA);
    float* hB   = (float*)(ws + o_hB);

    const int TB  = 256;
    const int gE  = (NEDGES + TB - 1) / TB;             // edge-parallel
    const int gN  = (NNODES + TB - 1) / TB;             // node-parallel
    const int gV4 = NNODES * (EDIM / 4) / TB;           // 4.8M/256 = 18750 (exact)
    const int gW  = ((NNODES / 16) + 7) / 8;            // 18750 waves, 8 waves/block

    // ---- build normalization + CSR (reused by all 3 layers) ----
    hipMemsetAsync(cnt, 0, (size_t)NNODES * 4, stream);
    lg_deg_kernel<<<gE, TB, 0, stream>>>(row, cnt);
    lg_dis_kernel<<<gN, TB, 0, stream>>>(cnt, dis);
    lg_scan_block<<<NBLK_SCAN, SCAN_B, 0, stream>>>(cnt, rptr, blk);
    lg_scan_blksums<<<1, 1, 0, stream>>>(blk);
    lg_scan_add<<<NBLK_SCAN, SCAN_B, 0, stream>>>(rptr, blk);
    hipMemsetAsync(cnt, 0, (size_t)NNODES * 4, stream); // reuse as fill counters
    lg_fill_kernel<<<gE, TB, 0, stream>>>(row, col, dis, rptr, cnt, ccol, cw);

    // ---- h0 = x, acc = x ----
    lg_init_kernel<<<gV4, TB, 0, stream>>>((const float4*)user, (const float4*)item,
                                           (float4*)hA, (float4*)out);

    // ---- layer 1: hA -> hB, acc += h1 (fused) ----
    lg_gather_kernel<true><<<gV4, TB, 0, stream>>>(rptr, ccol, cw, dis,
                                                   (const float4*)hA, (float4*)hB, (float4*)out);
    // ---- layer 2: hB -> hA, acc += h2 (fused) ----
    lg_gather_kernel<true><<<gV4, TB, 0, stream>>>(rptr, ccol, cw, dis,
                                                   (const float4*)hB, (float4*)hA, (float4*)out);
    // ---- layer 3: hA -> hB (no acc; epilogue consumes it) ----
    lg_gather_kernel<false><<<gV4, TB, 0, stream>>>(rptr, ccol, cw, dis,
                                                    (const float4*)hA, (float4*)hB, (float4*)out);

    // ---- out = 0.25*(acc + h3) via f32 WMMA chain ----
    lg_finalize_wmma<<<gW, TB, 0, stream>>>(out, hB, out);
}